// Cheb_GCNN_10L_54485955117440
// MI455X (gfx1250) — compile-verified
//
#include <hip/hip_runtime.h>
#include <hip/hip_bf16.h>
#include <cstddef>
#include <cstdint>

// ---------------------------------------------------------------------------
// Cheb-GCNN 10L for MI455X (gfx1250, wave32).
// fp32 WMMA (V_WMMA_F32_16X16X4_F32) for all dense matmuls; scatter-adds via
// global_atomic_add_f32 (L2-resident); BN stats via LDS ds_add_f32 + global
// atomics. W staging into LDS is double-buffered and uses the gfx1250 async
// global->LDS path (GLOBAL_LOAD_ASYNC_TO_LDS_B128 + s_wait_asynccnt) when the
// toolchain exposes the builtin.
// ---------------------------------------------------------------------------

#define N_NODES 50000
#define N_EDGES 600000
#define FDIM    128
#define NF      (N_NODES * FDIM)          // 6,400,000 floats per node buffer

typedef __attribute__((ext_vector_type(2))) float v2f;
typedef __attribute__((ext_vector_type(8))) float v8f;
typedef __attribute__((ext_vector_type(4))) int   v4i;

#if defined(__has_builtin)
#if __has_builtin(__builtin_amdgcn_global_load_async_to_lds_b128) && \
    __has_builtin(__builtin_amdgcn_s_wait_asynccnt)
#define HAVE_ASYNC_LDS 1
#endif
#endif

#ifdef HAVE_ASYNC_LDS
// generic->AS1 / generic->AS3 via integer (standard AMDGPU trick: generic LDS
// address low 32 bits == LDS byte offset; generic global == full VA).
// Builtin signature (from clang diagnostic): (int4 AS1*, int4 AS3*, imm, imm).
__device__ __forceinline__ void async_copy16(const float* gsrc, float* ldst) {
  __builtin_amdgcn_global_load_async_to_lds_b128(
      (__attribute__((address_space(1))) v4i*)(uintptr_t)gsrc,
      (__attribute__((address_space(3))) v4i*)(uint32_t)(uintptr_t)ldst,
      0, 0);
}
#endif

// ------------------------------- utilities ---------------------------------

__global__ void zero_f(float* __restrict__ p, int n) {
  int i = blockIdx.x * blockDim.x + threadIdx.x;
  if (i < n) p[i] = 0.0f;
}

__global__ void zero_f4(float4* __restrict__ p, int n4) {
  int i = blockIdx.x * blockDim.x + threadIdx.x;
  if (i < n4) p[i] = make_float4(0.f, 0.f, 0.f, 0.f);
}

// dst = -src  (initializes the Chebyshev recurrence target: T_k = -T_{k-2} + 2*L*T_{k-1})
__global__ void neg_copy_f4(const float4* __restrict__ src, float4* __restrict__ dst, int n4) {
  int i = blockIdx.x * blockDim.x + threadIdx.x;
  if (i < n4) {
    float4 v = src[i];
    dst[i] = make_float4(-v.x, -v.y, -v.z, -v.w);
  }
}

// ------------------------ edge normalization -------------------------------

__global__ void deg_accum(const int* __restrict__ row, const int* __restrict__ col,
                          const float* __restrict__ w, float* __restrict__ deg) {
  int e = blockIdx.x * blockDim.x + threadIdx.x;
  if (e < N_EDGES) {
    int r = row[e], c = col[e];
    if (r != c) atomicAdd(&deg[r], w[e]);
  }
}

// deg -> dis in place: dis = deg>0 ? rsqrt(deg) : 0
__global__ void deg_to_dis(float* __restrict__ deg) {
  int i = blockIdx.x * blockDim.x + threadIdx.x;
  if (i < N_NODES) {
    float d = deg[i];
    deg[i] = (d > 0.0f) ? rsqrtf(d) : 0.0f;
  }
}

__global__ void edge_norm(const int* __restrict__ row, const int* __restrict__ col,
                          const float* __restrict__ w, const float* __restrict__ dis,
                          float* __restrict__ nrm) {
  int e = blockIdx.x * blockDim.x + threadIdx.x;
  if (e < N_EDGES) {
    int r = row[e], c = col[e];
    nrm[e] = (r == c) ? 0.0f : (-dis[r] * w[e] * dis[c]);
  }
}

// ------------------------------ propagate ----------------------------------
// hdst[dst] += scale * norm[e] * hsrc[src]   (32 threads per edge, float4 each)

__global__ void scatter_prop(const int* __restrict__ srcIdx, const int* __restrict__ dstIdx,
                             const float* __restrict__ nrm,
                             const float* __restrict__ hsrc, float* __restrict__ hdst,
                             float scale) {
  long tid = (long)blockIdx.x * blockDim.x + threadIdx.x;
  long e = tid >> 5;
  if (e >= N_EDGES) return;
  float w = scale * nrm[e];
  if (w == 0.0f) return;               // self-loops / isolated endpoints
  int f = (int)(tid & 31) << 2;        // 0..124, float4-aligned column
  int s = srcIdx[e], d = dstIdx[e];
  const float4 v = *(const float4*)(hsrc + (size_t)s * FDIM + f);
  float* p = hdst + (size_t)d * FDIM + f;
  atomicAdd(p + 0, w * v.x);
  atomicAdd(p + 1, w * v.y);
  atomicAdd(p + 2, w * v.z);
  atomicAdd(p + 3, w * v.w);
}

// ------------------------------- ChebConv GEMM -----------------------------
// out[n,:] = relu( sum_k Tk[n,:] @ Wk + bias ), and accumulate per-column
// BN statistics (sum, sum of squares) into stats[0..127], stats[128..255].
//
// Workgroup: 256 threads = 8 waves. Tile: 128 rows x 128 cols.
// Wave w handles rows [blk*128 + w*16, +16) against all 8 column tiles.
// K = 4 segments (T0..T3) x 128 = 512, consumed 4 at a time by
// V_WMMA_F32_16X16X4_F32.  W staged 32x128 (16KB) per chunk into LDS,
// double-buffered: one barrier per chunk, next chunk's loads (async
// global->LDS when available) overlap the current chunk's 64 WMMAs.
//
// f32 16x4 A layout:  lane<16 -> {a.x=K0,a.y=K1}, lane>=16 -> {K2,K3}, M=lane%16
// f32 4x16 B layout:  mirrored (rows striped across lanes), N=lane%16
// f32 16x16 C layout: VGPR i -> M = i + 8*(lane>=16), N = lane%16

__global__ __launch_bounds__(256) void cheb_gemm(
    const float* __restrict__ T0, const float* __restrict__ T1,
    const float* __restrict__ T2, const float* __restrict__ T3,
    const float* __restrict__ W,      // [4][128][128] == [512][128] (one layer)
    const float* __restrict__ bias,   // [128]
    float* __restrict__ out,          // [N_NODES][128]
    float* __restrict__ stats)        // [256] = {sum[128], sumsq[128]}
{
  __shared__ float sW[2][32 * FDIM];  // 2 x 16 KB double-buffered W stage
  __shared__ float sSum[FDIM];
  __shared__ float sSq[FDIM];

  const int tid  = threadIdx.x;
  const int wave = tid >> 5;
  const int lane = tid & 31;
  const int ln16 = lane & 15;
  const int hi   = lane >> 4;         // 0: lanes 0-15, 1: lanes 16-31
  const int rowBase = blockIdx.x * 128 + wave * 16;
  const int arow = rowBase + ln16;    // A-matrix row for this lane

  if (tid < FDIM) { sSum[tid] = 0.0f; sSq[tid] = 0.0f; }

  v8f acc[8];
#pragma unroll
  for (int n = 0; n < 8; ++n) acc[n] = (v8f)0.0f;

  // stage W chunk kidx (rows [kidx*32, +32), 4096 floats) into sW[buf]:
  // thread t copies 64 contiguous bytes -> b128 global loads + b128 LDS stores.
#define STAGE_CHUNK(buf_, kidx_)                                              \
  do {                                                                        \
    const float* gsrc_ = W + (kidx_) * 32 * FDIM + tid * 16;                  \
    float* ldst_ = &sW[(buf_)][tid * 16];                                     \
    STAGE_IMPL(gsrc_, ldst_);                                                 \
  } while (0)

#ifdef HAVE_ASYNC_LDS
#define STAGE_IMPL(g_, l_)                                                    \
  do {                                                                        \
    async_copy16((g_) + 0,  (l_) + 0);                                        \
    async_copy16((g_) + 4,  (l_) + 4);                                        \
    async_copy16((g_) + 8,  (l_) + 8);                                        \
    async_copy16((g_) + 12, (l_) + 12);                                       \
  } while (0)
#define STAGE_WAIT() __builtin_amdgcn_s_wait_asynccnt(0)
#else
#define STAGE_IMPL(g_, l_)                                                    \
  do {                                                                        \
    float4 t0_ = *(const float4*)((g_) + 0);                                  \
    float4 t1_ = *(const float4*)((g_) + 4);                                  \
    float4 t2_ = *(const float4*)((g_) + 8);                                  \
    float4 t3_ = *(const float4*)((g_) + 12);                                 \
    *(float4*)((l_) + 0)  = t0_;                                              \
    *(float4*)((l_) + 4)  = t1_;                                              \
    *(float4*)((l_) + 8)  = t2_;                                              \
    *(float4*)((l_) + 12) = t3_;                                              \
  } while (0)
#define STAGE_WAIT() ((void)0)
#endif

  STAGE_CHUNK(0, 0);

  // 16 K-chunks of 32: chunk kidx covers global K [kidx*32, kidx*32+32)
  for (int kidx = 0; kidx < 16; ++kidx) {
    const float* Tsrc = (kidx < 8) ? ((kidx < 4) ? T0 : T1)
                                   : ((kidx < 12) ? T2 : T3);
    const int kk0 = (kidx & 3) * 32;  // offset inside the 128-wide segment

    STAGE_WAIT();
    __syncthreads();                  // chunk kidx resident; prev reads done
    if (kidx + 1 < 16) STAGE_CHUNK((kidx + 1) & 1, kidx + 1);

    const float* curW = sW[kidx & 1];
    const float* trow = Tsrc + (size_t)arow * FDIM + kk0 + 2 * hi;
#pragma unroll
    for (int kk = 0; kk < 32; kk += 4) {
      v2f a; a.x = 0.0f; a.y = 0.0f;
      if (arow < N_NODES) {
        v2f av = *(const v2f*)(trow + kk);   // single b64 load
        a = av;
      }
#pragma unroll
      for (int n = 0; n < 8; ++n) {
        v2f b;
        b.x = curW[(kk + 2 * hi)     * FDIM + n * 16 + ln16];
        b.y = curW[(kk + 2 * hi + 1) * FDIM + n * 16 + ln16];
        acc[n] = __builtin_amdgcn_wmma_f32_16x16x4_f32(
            false, a, false, b, (short)0, acc[n], false, false);
      }
    }
  }

  // Epilogue: bias + relu + store + BN stats (LDS reduce, then global atomic).
#pragma unroll
  for (int n = 0; n < 8; ++n) {
    const int col = n * 16 + ln16;
    const float bcol = bias[col];
    float psum = 0.0f, psq = 0.0f;
#pragma unroll
    for (int i = 0; i < 8; ++i) {
      int row = rowBase + 8 * hi + i;
      float v = acc[n][i] + bcol;
      v = v > 0.0f ? v : 0.0f;
      if (row < N_NODES) {
        out[(size_t)row * FDIM + col] = v;
        psum += v;
        psq  += v * v;
      }
    }
    atomicAdd(&sSum[col], psum);   // ds_add_f32
    atomicAdd(&sSq[col],  psq);
  }
  __syncthreads();
  if (tid < FDIM) {
    atomicAdd(&stats[tid],        sSum[tid]);
    atomicAdd(&stats[FDIM + tid], sSq[tid]);
  }
#undef STAGE_CHUNK
#undef STAGE_IMPL
#undef STAGE_WAIT
}

// --------------------------- batchnorm pieces ------------------------------

__global__ void bn_finalize(const float* __restrict__ stats,
                            const float* __restrict__ gamma,
                            const float* __restrict__ beta,
                            float* __restrict__ ss) {   // [256] = {scale, shift}
  int f = threadIdx.x;
  if (f < FDIM) {
    float inv_n = 1.0f / (float)N_NODES;
    float mean = stats[f] * inv_n;
    float var  = stats[FDIM + f] * inv_n - mean * mean;
    float s = gamma[f] * rsqrtf(var + 1e-5f);
    ss[f]        = s;
    ss[FDIM + f] = beta[f] - mean * s;
  }
}

__global__ void bn_apply(const float4* __restrict__ src, const float* __restrict__ ss,
                         float4* __restrict__ dst) {
  int i = blockIdx.x * blockDim.x + threadIdx.x;   // one float4 per thread
  if (i < NF / 4) {
    int f = (i & 31) << 2;                         // column of first element
    float4 v = src[i];
    float4 r;
    r.x = v.x * ss[f + 0] + ss[FDIM + f + 0];
    r.y = v.y * ss[f + 1] + ss[FDIM + f + 1];
    r.z = v.z * ss[f + 2] + ss[FDIM + f + 2];
    r.w = v.w * ss[f + 3] + ss[FDIM + f + 3];
    dst[i] = r;
  }
}

// --------------------------- final 128 -> 16 linear -------------------------

__global__ __launch_bounds__(256) void final_linear(
    const float* __restrict__ h, const float* __restrict__ lw,  // [128][16]
    const float* __restrict__ lb, float* __restrict__ out)      // [N_NODES][16]
{
  const int wave = threadIdx.x >> 5;
  const int lane = threadIdx.x & 31;
  const int ln16 = lane & 15;
  const int hi   = lane >> 4;
  const int rowBase = (blockIdx.x * 8 + wave) * 16;
  const int arow = rowBase + ln16;

  v8f acc = (v8f)0.0f;
  const float* hrow = h + (size_t)arow * FDIM + 2 * hi;
#pragma unroll 8
  for (int kk = 0; kk < FDIM; kk += 4) {
    v2f a; a.x = 0.0f; a.y = 0.0f;
    if (arow < N_NODES) {
      v2f av = *(const v2f*)(hrow + kk);
      a = av;
    }
    v2f b;
    b.x = lw[(kk + 2 * hi)     * 16 + ln16];
    b.y = lw[(kk + 2 * hi + 1) * 16 + ln16];
    acc = __builtin_amdgcn_wmma_f32_16x16x4_f32(
        false, a, false, b, (short)0, acc, false, false);
  }
  const float bc = lb[ln16];
#pragma unroll
  for (int i = 0; i < 8; ++i) {
    int row = rowBase + 8 * hi + i;
    if (row < N_NODES) out[(size_t)row * 16 + ln16] = acc[i] + bc;
  }
}

// ------------------------------ host driver --------------------------------

extern "C" void kernel_launch(void* const* d_in, const int* in_sizes, int n_in,
                              void* d_out, int out_size, void* d_ws, size_t ws_size,
                              hipStream_t stream) {
  const float* x      = (const float*)d_in[0];           // [50000,128]
  const int*   ei     = (const int*)  d_in[1];           // [2,600000]
  const float* ew     = (const float*)d_in[2];           // [600000]
  const float* conv_w = (const float*)d_in[3];           // [10,4,128,128]
  const float* conv_b = (const float*)d_in[4];           // [10,128]
  const float* gamma  = (const float*)d_in[5];           // [10,128]
  const float* beta   = (const float*)d_in[6];           // [10,128]
  const float* lw     = (const float*)d_in[7];           // [128,16]
  const float* lb     = (const float*)d_in[8];           // [16]
  float* out16 = (float*)d_out;                          // [50000,16]

  const int* rowI = ei;                // src
  const int* colI = ei + N_EDGES;      // dst

  // workspace layout (floats)
  float* ws    = (float*)d_ws;
  float* hbuf  = ws;                   // T0 for layers >= 1
  float* t1    = hbuf + NF;
  float* t2    = t1 + NF;
  float* t3    = t2 + NF;
  float* ob    = t3 + NF;              // conv output (pre-BN)
  float* nrm   = ob + NF;              // [600000]
  float* deg   = nrm + N_EDGES;        // [50000] (becomes dis in place)
  float* stats = deg + N_NODES;        // [256]
  float* ssbuf = stats + 256;          // [256]

  const int B = 256;
  const dim3 blk(B);
  const int gNodes  = (N_NODES + B - 1) / B;
  const int gEdges  = (N_EDGES + B - 1) / B;
  const int gNF4    = (NF / 4 + B - 1) / B;
  const int gScat   = (int)(((long)N_EDGES * 32 + B - 1) / B);
  const int gGemm   = (N_NODES + 127) / 128;

  // ---- edge normalization ----
  zero_f<<<gNodes, blk, 0, stream>>>(deg, N_NODES);
  deg_accum<<<gEdges, blk, 0, stream>>>(rowI, colI, ew, deg);
  deg_to_dis<<<gNodes, blk, 0, stream>>>(deg);
  edge_norm<<<gEdges, blk, 0, stream>>>(rowI, colI, ew, deg, nrm);

  // ---- 10 ChebConv + BN layers ----
  for (int layer = 0; layer < 10; ++layer) {
    const float* T0 = (layer == 0) ? x : hbuf;
    const float* W  = conv_w + (size_t)layer * 4 * FDIM * FDIM;
    const float* bI = conv_b + (size_t)layer * FDIM;

    // T1 = L^ T0
    zero_f4<<<gNF4, blk, 0, stream>>>((float4*)t1, NF / 4);
    scatter_prop<<<gScat, blk, 0, stream>>>(rowI, colI, nrm, T0, t1, 1.0f);
    // T2 = 2 L^ T1 - T0
    neg_copy_f4<<<gNF4, blk, 0, stream>>>((const float4*)T0, (float4*)t2, NF / 4);
    scatter_prop<<<gScat, blk, 0, stream>>>(rowI, colI, nrm, t1, t2, 2.0f);
    // T3 = 2 L^ T2 - T1
    neg_copy_f4<<<gNF4, blk, 0, stream>>>((const float4*)t1, (float4*)t3, NF / 4);
    scatter_prop<<<gScat, blk, 0, stream>>>(rowI, colI, nrm, t2, t3, 2.0f);

    // out = relu(sum_k Tk @ Wk + b), with BN stat accumulation
    zero_f<<<1, blk, 0, stream>>>(stats, 256);
    cheb_gemm<<<gGemm, blk, 0, stream>>>(T0, t1, t2, t3, W, bI, ob, stats);

    // BN -> hbuf
    bn_finalize<<<1, dim3(128), 0, stream>>>(stats, gamma + layer * FDIM,
                                             beta + layer * FDIM, ssbuf);
    bn_apply<<<gNF4, blk, 0, stream>>>((const float4*)ob, ssbuf, (float4*)hbuf);
  }

  // ---- final linear 128 -> 16 ----
  final_linear<<<gGemm, blk, 0, stream>>>(hbuf, lw, lb, out16);
}